// MMDLossFunction_29755533427519
// MI455X (gfx1250) — compile-verified
//
#include <hip/hip_runtime.h>
#include <hip/hip_bf16.h>

// CDNA5 (gfx1250) fused MMD-RBF kernel.
// Gram matrices computed with V_WMMA_F32_16X16X4_F32 (full fp32 matrix op),
// fused exp + hierarchical reduction, deterministic (no float atomics).

typedef __attribute__((ext_vector_type(2))) float v2f;
typedef __attribute__((ext_vector_type(8))) float v8f;

#define TILE 128          // output tile per block (128x128)
#define KCH  16           // K panel per LDS stage
#define LSTR 20           // padded LDS row stride in floats (20%64 -> conflict-free, 16B aligned)

__global__ __launch_bounds__(256) void row_sqnorm_kernel(const float* __restrict__ A,
                                                         float* __restrict__ sq,
                                                         int N, int D) {
    int row = blockIdx.x * blockDim.x + threadIdx.x;
    if (row >= N) return;
    const float4* p = reinterpret_cast<const float4*>(A + (size_t)row * D);
    float s = 0.f;
    int n4 = D >> 2;
    for (int k = 0; k < n4; ++k) {
        float4 v = p[k];
        s += v.x * v.x + v.y * v.y + v.z * v.z + v.w * v.w;
    }
    sq[row] = s;
}

// One block computes a 128x128 tile of exp(-0.5*d2(a_i,b_j)) and emits its sum.
// 8 waves; each wave owns a 32x64 sub-tile = 2x4 grid of 16x16 WMMA accumulators.
__global__ __launch_bounds__(256) void gram_exp_sum_kernel(
    const float* __restrict__ A, const float* __restrict__ B,
    const float* __restrict__ sqA, const float* __restrict__ sqB,
    float* __restrict__ partial, int N, int D)
{
    extern __shared__ float smem[];
    float* As = smem;                  // TILE x LSTR
    float* Bs = smem + TILE * LSTR;    // TILE x LSTR

    const int tid  = threadIdx.x;
    const int lane = tid & 31;
    const int wave = tid >> 5;
    const int half = lane >> 4;   // 0: lanes 0-15, 1: lanes 16-31
    const int l16  = lane & 15;

    const int rowBase = blockIdx.y * TILE;
    const int colBase = blockIdx.x * TILE;

    const int wrow = (wave >> 1) * 32;  // 0,32,64,96
    const int wcol = (wave & 1) * 64;   // 0,64

    v8f zero = {};
    v8f acc[2][4];
    #pragma unroll
    for (int i = 0; i < 2; ++i)
        #pragma unroll
        for (int j = 0; j < 4; ++j) acc[i][j] = zero;

    for (int kk = 0; kk < D; kk += KCH) {
        // Cooperative panel load: 128 rows x 16 floats per panel = 512 float4 each.
        for (int t = tid; t < TILE * 4; t += 256) {
            int r  = t >> 2;
            int c4 = (t & 3) * 4;
            float4 va = *reinterpret_cast<const float4*>(A + (size_t)(rowBase + r) * D + kk + c4);
            float4 vb = *reinterpret_cast<const float4*>(B + (size_t)(colBase + r) * D + kk + c4);
            *reinterpret_cast<float4*>(&As[r * LSTR + c4]) = va;
            *reinterpret_cast<float4*>(&Bs[r * LSTR + c4]) = vb;
        }
        __syncthreads();

        #pragma unroll
        for (int kc = 0; kc < KCH; kc += 4) {
            // Fragment layout (16x4 fp32 A / 4x16 fp32 B):
            // lane half 0 holds K=kc+0..1, half 1 holds K=kc+2..3, one matrix row/col per l16.
            const int koff = kc + 2 * half;
            v2f afrag[2], bfrag[4];
            #pragma unroll
            for (int i = 0; i < 2; ++i)
                afrag[i] = *reinterpret_cast<const v2f*>(&As[(wrow + i * 16 + l16) * LSTR + koff]);
            #pragma unroll
            for (int j = 0; j < 4; ++j)
                bfrag[j] = *reinterpret_cast<const v2f*>(&Bs[(wcol + j * 16 + l16) * LSTR + koff]);

            #pragma unroll
            for (int i = 0; i < 2; ++i)
                #pragma unroll
                for (int j = 0; j < 4; ++j)
                    acc[i][j] = __builtin_amdgcn_wmma_f32_16x16x4_f32(
                        /*neg_a=*/false, afrag[i],
                        /*neg_b=*/false, bfrag[j],
                        /*c_mod=*/(short)0, acc[i][j],
                        /*reuse_a=*/false, /*reuse_b=*/false);
        }
        __syncthreads();
    }

    // Epilogue: d2 = max(sq_a + sq_b - 2*g, 0); sum exp(-0.5*d2).
    // C/D layout: VGPR v, lane -> row M = v + 8*half, col N = l16.
    float lsum = 0.f;
    #pragma unroll
    for (int i = 0; i < 2; ++i) {
        const int i0 = rowBase + wrow + i * 16 + half * 8;
        #pragma unroll
        for (int j = 0; j < 4; ++j) {
            const int jj = colBase + wcol + j * 16 + l16;
            const float sb = sqB[jj];
            #pragma unroll
            for (int v = 0; v < 8; ++v) {
                float g  = acc[i][j][v];
                float sa = sqA[i0 + v];
                float d2 = fmaxf(sa + sb - 2.f * g, 0.f);
                lsum += __expf(-0.5f * d2);
            }
        }
    }

    // wave32 reduction, then cross-wave via LDS (deterministic).
    #pragma unroll
    for (int off = 16; off > 0; off >>= 1)
        lsum += __shfl_down(lsum, off, 32);
    __syncthreads();
    if (lane == 0) smem[wave] = lsum;
    __syncthreads();
    if (tid == 0) {
        float s = 0.f;
        #pragma unroll
        for (int w = 0; w < 8; ++w) s += smem[w];
        partial[blockIdx.y * gridDim.x + blockIdx.x] = s;
    }
}

__global__ __launch_bounds__(256) void finalize_kernel(const float* __restrict__ partial, int nper,
                                                       const float* __restrict__ avg_step,
                                                       float* __restrict__ out, float invNN) {
    __shared__ float red[3][8];
    const int tid = threadIdx.x;
    const int lane = tid & 31, wave = tid >> 5;
    float s[3] = {0.f, 0.f, 0.f};
    for (int c = 0; c < 3; ++c)
        for (int k = tid; k < nper; k += 256)
            s[c] += partial[c * nper + k];
    #pragma unroll
    for (int off = 16; off > 0; off >>= 1)
        for (int c = 0; c < 3; ++c) s[c] += __shfl_down(s[c], off, 32);
    if (lane == 0)
        for (int c = 0; c < 3; ++c) red[c][wave] = s[c];
    __syncthreads();
    if (tid == 0) {
        float t[3];
        for (int c = 0; c < 3; ++c) {
            t[c] = 0.f;
            for (int w = 0; w < 8; ++w) t[c] += red[c][w];
        }
        const float XX = t[0] * invNN;
        const float XY = t[1] * invNN;
        const float YY = t[2] * invNN;
        const float mmd = XX + YY - 2.f * XY;
        const float a = avg_step[0];
        const float loss = mmd + (fmaxf(1.f, a) - 1.f) * 0.002f;
        out[0] = loss;  // loss
        out[1] = mmd;   // mmd
    }
}

extern "C" void kernel_launch(void* const* d_in, const int* in_sizes, int n_in,
                              void* d_out, int out_size, void* d_ws, size_t ws_size,
                              hipStream_t stream) {
    const float* x        = (const float*)d_in[0];
    const float* y        = (const float*)d_in[1];
    const float* avg_step = (const float*)d_in[2];
    float* out = (float*)d_out;

    const int D = 24 * 8;                // 192, flattened feature dim
    const int N = in_sizes[0] / D;       // 4096
    const int M = in_sizes[1] / D;       // 4096

    float* sqA  = (float*)d_ws;          // N floats
    float* sqB  = sqA + N;               // M floats
    float* part = sqB + M;               // 3 * (N/TILE)^2 floats

    const int tiles = N / TILE;          // 32
    const int nper  = tiles * tiles;     // 1024 partials per Gram matrix

    row_sqnorm_kernel<<<(N + 255) / 256, 256, 0, stream>>>(x, sqA, N, D);
    row_sqnorm_kernel<<<(M + 255) / 256, 256, 0, stream>>>(y, sqB, M, D);

    dim3 grid(tiles, tiles);
    const size_t lds_bytes = 2u * TILE * LSTR * sizeof(float);  // 20.5 KB
    gram_exp_sum_kernel<<<grid, 256, lds_bytes, stream>>>(x, x, sqA, sqA, part,            N, D);
    gram_exp_sum_kernel<<<grid, 256, lds_bytes, stream>>>(x, y, sqA, sqB, part + nper,     N, D);
    gram_exp_sum_kernel<<<grid, 256, lds_bytes, stream>>>(y, y, sqB, sqB, part + 2 * nper, N, D);

    finalize_kernel<<<1, 256, 0, stream>>>(part, nper, avg_step, out,
                                           1.f / ((float)N * (float)N));
}